// CustomSPNModel_48945447305702
// MI455X (gfx1250) — compile-verified
//
#include <hip/hip_runtime.h>
#include <hip/hip_bf16.h>
#include <cfloat>

#define LRELU_SLOPE 0.2f

typedef __attribute__((ext_vector_type(2))) float v2f;
typedef __attribute__((ext_vector_type(4))) float v4f;
typedef __attribute__((ext_vector_type(8))) float v8f;

static inline int ceildiv(int a, int b) { return (a + b - 1) / b; }

// ---------------------------------------------------------------------------
// float atomic max via monotone integer ordering (init buffer to -FLT_MAX)
// ---------------------------------------------------------------------------
static __device__ __forceinline__ void atomicMaxF(float* addr, float v) {
    int iv = __float_as_int(v);
    if (iv >= 0) atomicMax((int*)addr, iv);
    else         atomicMin((unsigned int*)addr, (unsigned int)iv);
}

// ---------------------------------------------------------------------------
// fill kernels (graph-capture safe, handle -FLT_MAX)
// ---------------------------------------------------------------------------
__global__ void fillk(float* __restrict__ p, float v, int n) {
    int t = blockIdx.x * blockDim.x + threadIdx.x;
    if (t < n) p[t] = v;
}
__global__ void fillk4(v4f* __restrict__ p, float v, int n4) {
    int t = blockIdx.x * blockDim.x + threadIdx.x;
    if (t < n4) { v4f w = {v, v, v, v}; p[t] = w; }
}

// ---------------------------------------------------------------------------
// FP32 WMMA GEMM: C[M,N] = act(A[M,K] @ B[K,N] + bias)
// Wave computes 16 rows x 64 cols via 4x V_WMMA_F32_16X16X4_F32.
// Block = 128 threads (4 waves) -> 64 rows. grid = (ceil(M/64), N/64).
// Requires: M % 16 == 0, N % 64 == 0, K % 4 == 0 (true for all call sites).
// ACT: 0 = none, 1 = ELU, 2 = ReLU
// ---------------------------------------------------------------------------
template <int ACT>
__global__ __launch_bounds__(128)
void gemm_wmma_f32(const float* __restrict__ A, const float* __restrict__ B,
                   const float* __restrict__ bias, float* __restrict__ C,
                   int M, int K, int N) {
    const int lane = threadIdx.x & 31;
    const int wave = threadIdx.x >> 5;
    const int row0 = (blockIdx.x * 4 + wave) * 16;
    const int col0 = blockIdx.y * 64;
    if (row0 >= M) return;                 // wave-uniform: EXEC stays all-1s
    const int half = lane >> 4;            // 0 or 1 (K sub-pair select)
    const int l16  = lane & 15;
    const int m    = row0 + l16;           // A row held by this lane

    // strength-reduced running pointers (avoid per-iter 64-bit multiplies)
    const float* Aptr = A + (size_t)m * K + 2 * half;
    const float* Bptr = B + (size_t)(2 * half) * N + col0 + l16;
    const size_t Bstep = (size_t)4 * N;

    v8f acc[4] = {};
    for (int k = 0; k < K; k += 4) {
        // A 16x4 tile: lane(l<16) holds A[m][k+2h], A[m][k+2h+1]
        v2f a = *(const v2f*)Aptr;
        __builtin_prefetch(Aptr + 64, 0, 0);   // stream A ~256B ahead
        Aptr += 4;
        // B 4x16 tiles: vgpr v, lane l -> B[k+2h+v][col0 + j*16 + l16]
        const float* B0 = Bptr;
        const float* B1 = Bptr + N;
        Bptr += Bstep;
        v2f b[4];
#pragma unroll
        for (int j = 0; j < 4; ++j) { b[j].x = B0[j * 16]; b[j].y = B1[j * 16]; }
#pragma unroll
        for (int j = 0; j < 4; ++j) {
            acc[j] = __builtin_amdgcn_wmma_f32_16x16x4_f32(
                false, a, false, b[j], (short)0, acc[j], false, false);
        }
    }
    // C/D layout: vgpr r, lane l -> C[row0 + r + 8*(l>>4)][col0 + j*16 + (l&15)]
#pragma unroll
    for (int j = 0; j < 4; ++j) {
        const int n = col0 + j * 16 + l16;
        const float bv = bias ? bias[n] : 0.0f;
#pragma unroll
        for (int r = 0; r < 8; ++r) {
            const int mm = row0 + r + 8 * half;
            float v = acc[j][r] + bv;
            if (ACT == 1) v = (v > 0.0f) ? v : (__expf(v) - 1.0f);  // ELU
            if (ACT == 2) v = (v > 0.0f) ? v : 0.0f;                // ReLU
            C[(size_t)mm * N + n] = v;
        }
    }
}

// ---------------------------------------------------------------------------
// GATv2 stage 1: per-edge attention logits + segment max.
// Wave per edge; C = 64 fixed, H runtime (4 or 1). Lane covers float pair
// idx = h*64 + 2*lane (+1) via b64 loads. Self loops appended:
// e in [Ereal, Etot) -> s = d = e - Ereal.
// ---------------------------------------------------------------------------
__global__ __launch_bounds__(256)
void gat_logits(const float* __restrict__ XL, const float* __restrict__ XR,
                const float* __restrict__ att, const int* __restrict__ src,
                const int* __restrict__ dst, int Ereal, int Etot, int H,
                float* __restrict__ logits, float* __restrict__ mbuf) {
    const int e = blockIdx.x * 8 + (threadIdx.x >> 5);
    if (e >= Etot) return;
    const int lane = threadIdx.x & 31;
    int s, d;
    if (e < Ereal) { s = src[e]; d = dst[e]; } else { s = e - Ereal; d = s; }
    const int HC = H * 64;
    const float* xl = XL + (size_t)s * HC;
    const float* xr = XR + (size_t)d * HC;
    for (int h = 0; h < H; ++h) {
        const v2f xlv = ((const v2f*)(xl + h * 64))[lane];
        const v2f xrv = ((const v2f*)(xr + h * 64))[lane];
        const v2f av  = ((const v2f*)(att + h * 64))[lane];
        float v0 = xlv.x + xrv.x; v0 = (v0 > 0.0f) ? v0 : LRELU_SLOPE * v0;
        float v1 = xlv.y + xrv.y; v1 = (v1 > 0.0f) ? v1 : LRELU_SLOPE * v1;
        float p = v0 * av.x + v1 * av.y;
        for (int off = 16; off > 0; off >>= 1) p += __shfl_xor(p, off, 32);
        if (lane == 0) {
            logits[(size_t)e * H + h] = p;
            atomicMaxF(&mbuf[(size_t)d * H + h], p);
        }
    }
}

// ---------------------------------------------------------------------------
// GATv2 stage 2: p = exp(logit - m[d]) (in place), z[d] += p
// ---------------------------------------------------------------------------
__global__ void gat_expsum(float* __restrict__ logits, const float* __restrict__ mbuf,
                           float* __restrict__ zbuf, const int* __restrict__ dst,
                           int Ereal, int Etot, int H) {
    const int t = blockIdx.x * blockDim.x + threadIdx.x;
    if (t >= Etot * H) return;
    const int e = t / H, h = t - e * H;
    const int d = (e < Ereal) ? dst[e] : (e - Ereal);
    const float p = __expf(logits[t] - mbuf[(size_t)d * H + h]);
    logits[t] = p;
    atomicAdd(&zbuf[(size_t)d * H + h], p);
}

// ---------------------------------------------------------------------------
// GATv2 stage 3: acc[d] += (p/z[d]) * XL[s]   (wave per edge, b64 gathers)
// ---------------------------------------------------------------------------
__global__ __launch_bounds__(256)
void gat_scatter(const float* __restrict__ XL, const float* __restrict__ p,
                 const float* __restrict__ zbuf, const int* __restrict__ src,
                 const int* __restrict__ dst, int Ereal, int Etot, int H,
                 float* __restrict__ acc) {
    const int e = blockIdx.x * 8 + (threadIdx.x >> 5);
    if (e >= Etot) return;
    const int lane = threadIdx.x & 31;
    int s, d;
    if (e < Ereal) { s = src[e]; d = dst[e]; } else { s = e - Ereal; d = s; }
    const int HC = H * 64;
    const float* xl = XL + (size_t)s * HC;
    float* ad = acc + (size_t)d * HC;
    for (int h = 0; h < H; ++h) {
        const float alpha = p[(size_t)e * H + h] / zbuf[(size_t)d * H + h];
        const v2f xlv = ((const v2f*)(xl + h * 64))[lane];
        float* base = ad + h * 64 + 2 * lane;
        atomicAdd(base + 0, alpha * xlv.x);
        atomicAdd(base + 1, alpha * xlv.y);
    }
}

// ---------------------------------------------------------------------------
// out = act(acc + bias), vectorized x4 (ACT: 0 none, 1 ELU); total % 4 == 0
// ---------------------------------------------------------------------------
template <int ACT>
__global__ void finalize_k4(const v4f* __restrict__ acc, const float* __restrict__ bias,
                            v4f* __restrict__ out, int total4, int cols) {
    const int t = blockIdx.x * blockDim.x + threadIdx.x;
    if (t >= total4) return;
    v4f v = acc[t];
    const int c = (t * 4) % cols;           // cols % 4 == 0 -> same row, contiguous
    const v4f bv = *(const v4f*)(bias + c);
#pragma unroll
    for (int i = 0; i < 4; ++i) {
        float w = v[i] + bv[i];
        if (ACT == 1) w = (w > 0.0f) ? w : (__expf(w) - 1.0f);
        v[i] = w;
    }
    out[t] = v;
}

// ---------------------------------------------------------------------------
// CRF pairwise message: msg[dst[e]] += edge_repr[e] @ crf_wp  (wave per edge)
// Lane covers k = 2*lane, 2*lane+1 via b64 loads.
// ---------------------------------------------------------------------------
__global__ __launch_bounds__(256)
void crf_msg(const float* __restrict__ edge_repr, const float* __restrict__ wp,
             const int* __restrict__ dst, int E, float* __restrict__ msg) {
    const int e = blockIdx.x * 8 + (threadIdx.x >> 5);
    if (e >= E) return;
    const int lane = threadIdx.x & 31;
    const v2f er = ((const v2f*)(edge_repr + (size_t)e * 64))[lane];
    const v2f w0 = ((const v2f*)wp)[2 * lane + 0];   // wp[4l+0], wp[4l+1]
    const v2f w1 = ((const v2f*)wp)[2 * lane + 1];   // wp[4l+2], wp[4l+3]
    float v0 = er.x * w0.x + er.y * w1.x;
    float v1 = er.x * w0.y + er.y * w1.y;
    for (int off = 16; off > 0; off >>= 1) {
        v0 += __shfl_xor(v0, off, 32);
        v1 += __shfl_xor(v1, off, 32);
    }
    if (lane == 0) {
        const int d = dst[e];
        atomicAdd(&msg[d * 2 + 0], v0);
        atomicAdd(&msg[d * 2 + 1], v1);
    }
}

// ---------------------------------------------------------------------------
// CRF head: out = log_softmax(node_repr @ wu + bu + msg)
// ---------------------------------------------------------------------------
__global__ void crf_final(const float* __restrict__ node_repr, const float* __restrict__ wu,
                          const float* __restrict__ bu, const float* __restrict__ msg,
                          float* __restrict__ out, int Nn) {
    const int n = blockIdx.x * blockDim.x + threadIdx.x;
    if (n >= Nn) return;
    const float* x = node_repr + (size_t)n * 64;
    float o0 = bu[0] + msg[n * 2 + 0];
    float o1 = bu[1] + msg[n * 2 + 1];
#pragma unroll 4
    for (int k = 0; k < 64; ++k) {
        const float xv = x[k];
        o0 += xv * wu[k * 2 + 0];
        o1 += xv * wu[k * 2 + 1];
    }
    const float mx = fmaxf(o0, o1);
    const float lse = mx + __logf(__expf(o0 - mx) + __expf(o1 - mx));
    out[n * 2 + 0] = o0 - lse;
    out[n * 2 + 1] = o1 - lse;
}

// ---------------------------------------------------------------------------
// Proxy head final: out = ph @ px_w2 + px_b2
// ---------------------------------------------------------------------------
__global__ void proxy_final(const float* __restrict__ ph, const float* __restrict__ w2,
                            const float* __restrict__ b2, float* __restrict__ out, int Nn) {
    const int n = blockIdx.x * blockDim.x + threadIdx.x;
    if (n >= Nn) return;
    const float* x = ph + (size_t)n * 128;
    float o0 = b2[0], o1 = b2[1];
#pragma unroll 4
    for (int k = 0; k < 128; ++k) {
        const float xv = x[k];
        o0 += xv * w2[k * 2 + 0];
        o1 += xv * w2[k * 2 + 1];
    }
    out[n * 2 + 0] = o0;
    out[n * 2 + 1] = o1;
}

// ---------------------------------------------------------------------------
// Host helpers
// ---------------------------------------------------------------------------
static void fill(hipStream_t st, float* p, float v, int n) {
    if ((n & 3) == 0)
        hipLaunchKernelGGL(fillk4, dim3(ceildiv(n / 4, 256)), dim3(256), 0, st, (v4f*)p, v, n / 4);
    else
        hipLaunchKernelGGL(fillk, dim3(ceildiv(n, 256)), dim3(256), 0, st, p, v, n);
}

static void gemm(hipStream_t st, const float* A, const float* B, const float* bias,
                 float* C, int M, int K, int N, int act) {
    dim3 g(ceildiv(M, 64), N / 64), b(128);
    if (act == 1)      hipLaunchKernelGGL(gemm_wmma_f32<1>, g, b, 0, st, A, B, bias, C, M, K, N);
    else if (act == 2) hipLaunchKernelGGL(gemm_wmma_f32<2>, g, b, 0, st, A, B, bias, C, M, K, N);
    else               hipLaunchKernelGGL(gemm_wmma_f32<0>, g, b, 0, st, A, B, bias, C, M, K, N);
}

// Runs a full GATv2 layer *after* XL/XR GEMMs. XRacc doubles as the scatter
// accumulator (XR is dead after gat_logits). out may alias XRacc (elementwise).
static void gat_rest(hipStream_t st, const float* XL, float* XRacc,
                     const float* att, const float* bias,
                     const int* src, const int* dst,
                     int Ereal, int Nn, int H,
                     float* LG, float* MB, float* ZB,
                     float* out, int act) {
    const int Etot = Ereal + Nn;
    const int HC = H * 64;
    const int total = Nn * HC;              // multiple of 4 at all call sites
    fill(st, MB, -FLT_MAX, Nn * H);
    fill(st, ZB, 0.0f, Nn * H);
    hipLaunchKernelGGL(gat_logits, dim3(ceildiv(Etot, 8)), dim3(256), 0, st,
                       XL, XRacc, att, src, dst, Ereal, Etot, H, LG, MB);
    hipLaunchKernelGGL(gat_expsum, dim3(ceildiv(Etot * H, 256)), dim3(256), 0, st,
                       LG, MB, ZB, dst, Ereal, Etot, H);
    fill(st, XRacc, 0.0f, total);
    hipLaunchKernelGGL(gat_scatter, dim3(ceildiv(Etot, 8)), dim3(256), 0, st,
                       XL, LG, ZB, src, dst, Ereal, Etot, H, XRacc);
    if (act == 1)
        hipLaunchKernelGGL(finalize_k4<1>, dim3(ceildiv(total / 4, 256)), dim3(256), 0, st,
                           (const v4f*)XRacc, bias, (v4f*)out, total / 4, HC);
    else
        hipLaunchKernelGGL(finalize_k4<0>, dim3(ceildiv(total / 4, 256)), dim3(256), 0, st,
                           (const v4f*)XRacc, bias, (v4f*)out, total / 4, HC);
}

extern "C" void kernel_launch(void* const* d_in, const int* in_sizes, int n_in,
                              void* d_out, int out_size, void* d_ws, size_t ws_size,
                              hipStream_t stream) {
    (void)n_in; (void)out_size;
    const int F = 64, HID = 64, HEADS = 4;
    const int N = in_sizes[0] / F;   // 20000
    const int E = in_sizes[2] / 2;   // 320000

    const float* x         = (const float*)d_in[0];
    const float* edge_type = (const float*)d_in[1];
    const int*   src       = (const int*)d_in[2];
    const int*   dst       = src + E;
    const float* n1_wl = (const float*)d_in[3];
    const float* n1_wr = (const float*)d_in[4];
    const float* n1_att = (const float*)d_in[5];
    const float* n1_b  = (const float*)d_in[6];
    const float* n2_wl = (const float*)d_in[7];
    const float* n2_wr = (const float*)d_in[8];
    const float* n2_att = (const float*)d_in[9];
    const float* n2_b  = (const float*)d_in[10];
    const float* e1_wl = (const float*)d_in[11];
    const float* e1_wr = (const float*)d_in[12];
    const float* e1_att = (const float*)d_in[13];
    const float* e1_b  = (const float*)d_in[14];
    const float* e2_wl = (const float*)d_in[15];
    const float* e2_wr = (const float*)d_in[16];
    const float* e2_att = (const float*)d_in[17];
    const float* e2_b  = (const float*)d_in[18];
    const float* crf_wu = (const float*)d_in[19];
    const float* crf_bu = (const float*)d_in[20];
    const float* crf_wp = (const float*)d_in[21];
    const float* px_w1 = (const float*)d_in[22];
    const float* px_b1 = (const float*)d_in[23];
    const float* px_w2 = (const float*)d_in[24];
    const float* px_b2 = (const float*)d_in[25];

    // ----- workspace arena (floats); XR buffers alias the scatter accumulators
    float* ws = (float*)d_ws;
    size_t o = 0;
    float* P0  = ws + o; o += (size_t)E * 256;       // XL (layer-1 width)
    float* P1  = ws + o; o += (size_t)E * 256;       // XR1 -> ACC1 -> h / g
    float* P2  = ws + o; o += (size_t)E * 64;        // XL2
    float* P3  = ws + o; o += (size_t)E * 64;        // XR2 -> ACC2 -> edge_repr
    float* LG  = ws + o; o += (size_t)2 * E * HEADS; // logits / p
    float* MB  = ws + o; o += (size_t)E * HEADS;     // segment max
    float* ZB  = ws + o; o += (size_t)E * HEADS;     // segment sum
    float* NR  = ws + o; o += (size_t)N * HID;       // node_repr
    float* PHB = ws + o; o += (size_t)N * 128;       // proxy hidden
    float* MSG = ws + o; o += (size_t)N * 2;         // crf messages
    if (ws_size < o * sizeof(float)) return;         // scratch too small: bail

    // ----- node GNN: GATv2(H=4, concat) -> ELU -> GATv2(H=1)
    gemm(stream, x, n1_wl, nullptr, P0, N, F, 256, 0);
    gemm(stream, x, n1_wr, nullptr, P1, N, F, 256, 0);
    gat_rest(stream, P0, P1, n1_att, n1_b, src, dst, E, N, 4, LG, MB, ZB, P1, 1); // h in P1
    gemm(stream, P1, n2_wl, nullptr, P2, N, 256, 64, 0);
    gemm(stream, P1, n2_wr, nullptr, P3, N, 256, 64, 0);
    gat_rest(stream, P2, P3, n2_att, n2_b, src, dst, E, N, 1, LG, MB, ZB, NR, 0); // node_repr

    // ----- edge GNN over edge features (segments over E rows, self loop per row)
    gemm(stream, edge_type, e1_wl, nullptr, P0, E, F, 256, 0);
    gemm(stream, edge_type, e1_wr, nullptr, P1, E, F, 256, 0);
    gat_rest(stream, P0, P1, e1_att, e1_b, src, dst, E, E, 4, LG, MB, ZB, P1, 1); // g in P1
    gemm(stream, P1, e2_wl, nullptr, P2, E, 256, 64, 0);
    gemm(stream, P1, e2_wr, nullptr, P3, E, 256, 64, 0);
    gat_rest(stream, P2, P3, e2_att, e2_b, src, dst, E, E, 1, LG, MB, ZB, P3, 0); // edge_repr in P3

    // ----- CRF head
    float* outF = (float*)d_out;
    fill(stream, MSG, 0.0f, N * 2);
    hipLaunchKernelGGL(crf_msg, dim3(ceildiv(E, 8)), dim3(256), 0, stream, P3, crf_wp, dst, E, MSG);
    hipLaunchKernelGGL(crf_final, dim3(ceildiv(N, 256)), dim3(256), 0, stream,
                       NR, crf_wu, crf_bu, MSG, outF, N);

    // ----- proxy head
    gemm(stream, NR, px_w1, px_b1, PHB, N, HID, 128, 2);
    hipLaunchKernelGGL(proxy_final, dim3(ceildiv(N, 256)), dim3(256), 0, stream,
                       PHB, px_w2, px_b2, outF + (size_t)N * 2, N);
}